// CannyEdge_83245056131622
// MI455X (gfx1250) — compile-verified
//
#include <hip/hip_runtime.h>
#include <stdint.h>

// ---------------------------------------------------------------------------
// Canny (cv2 defaults: L1 grad, thresholds 100/200, aperture 3) on MI455X.
// Pipeline: init -> gray/sobel/nms/threshold -> 10x (10-step in-LDS hysteresis
// dilation, tiles DMA'd via the CDNA5 Tensor Data Mover) -> finalize.
// Byte masks live in 16-px zero-padded 2080x2080 buffers in d_ws (~13MB),
// kept L2-resident (192MB L2); the one-shot 48MB img read and 48MB out write
// use non-temporal hints so streaming traffic doesn't evict the masks.
// ---------------------------------------------------------------------------

#define IMG_H 2048
#define IMG_W 2048
#define PAD   16
#define PP    (IMG_W + 2 * PAD)          // 2080 padded pitch (bytes)
#define PBYTES ((size_t)PP * PP)         // 4,326,400 bytes per mask buffer

typedef __attribute__((ext_vector_type(4))) unsigned int v4u;
typedef __attribute__((ext_vector_type(8))) int          v8i;
typedef __attribute__((ext_vector_type(4))) int          v4i;
typedef __attribute__((ext_vector_type(4))) float        v4f;

// --- TDM: 2D byte-tile load (tile_w x tile_h) from a 2080x2080 byte tensor
// into LDS at lds_addr. Descriptor per CDNA5 ISA ch.8 (D# groups 0/1).
__device__ __forceinline__ void tdm_load_tile_2d(unsigned lds_addr,
                                                 const unsigned char* gbase,
                                                 long long byteoff,
                                                 int tile_w, int tile_h) {
  unsigned long long ga = (unsigned long long)(uintptr_t)gbase +
                          (unsigned long long)byteoff;
  v4u g0;
  g0[0] = 1u;                                        // count=1, user mode
  g0[1] = lds_addr;                                  // LDS byte address
  g0[2] = (unsigned)(ga & 0xFFFFFFFFull);            // global_addr[31:0]
  g0[3] = (unsigned)((ga >> 32) & 0x01FFFFFFull)     // global_addr[56:32]
          | (2u << 30);                              // type = 2 (image)
  v8i g1;
  g1[0] = 0;                                         // wgmask=0, data_size=1B
  g1[1] = (int)(((unsigned)PP & 0xFFFFu) << 16);     // tensor_dim0[15:0]
  g1[2] = (int)((((unsigned)PP >> 16) & 0xFFFFu)     // tensor_dim0[31:16]
          | (((unsigned)PP & 0xFFFFu) << 16));       // tensor_dim1[15:0]
  g1[3] = (int)((((unsigned)PP >> 16) & 0xFFFFu)     // tensor_dim1[31:16]
          | (((unsigned)tile_w & 0xFFFFu) << 16));   // tile_dim0
  g1[4] = (int)((unsigned)tile_h & 0xFFFFu);         // tile_dim1 (tile_dim2=0)
  g1[5] = (int)(unsigned)PP;                         // tensor_dim0_stride[31:0]
  g1[6] = 0;                                         // stride0 hi / stride1 lo
  g1[7] = 0;
  v4i gz = {0, 0, 0, 0};
#if defined(__clang_major__) && (__clang_major__ >= 23)
  v8i gz8 = {0, 0, 0, 0, 0, 0, 0, 0};
  __builtin_amdgcn_tensor_load_to_lds(g0, g1, gz, gz, gz8, 0);
#else
  __builtin_amdgcn_tensor_load_to_lds(g0, g1, gz, gz, 0);
#endif
}

// --- Kernel 0: zero the padded workspace buffers (deterministic each call).
// PBYTES % 16 == 0, so uint4 stores cover the buffers exactly.
__global__ void canny_init_kernel(v4u* __restrict__ p, int nvec) {
  int i = blockIdx.x * blockDim.x + threadIdx.x;
  int stride = gridDim.x * blockDim.x;
  v4u z = {0u, 0u, 0u, 0u};
  for (; i < nvec; i += stride) p[i] = z;
}

// --- Kernel 1: fused gray -> Sobel -> L1 magnitude -> NMS -> double threshold.
// 32x32 output tile per block; gray in LDS with reflect-101 halo 2; magnitude
// on a 34x34 ring zero-padded outside the image (matches jnp.pad(mag, 1)).
// img is read exactly once -> non-temporal loads keep it out of L2.
__global__ __launch_bounds__(256) void canny_thresh_kernel(
    const float* __restrict__ img,
    unsigned char* __restrict__ edgesPad,
    unsigned char* __restrict__ weakPad) {
  __shared__ float sg[36 * 36];
  __shared__ float sm[34 * 34];
  const int tid = threadIdx.x;
  const int ox = blockIdx.x * 32, oy = blockIdx.y * 32;
  const long long HW = (long long)IMG_H * IMG_W;

  // Stage A: gray tile (36x36) with reflect-101 borders, scaled to [0,255].
  for (int i = tid; i < 36 * 36; i += 256) {
    int gy = i / 36, gx = i % 36;
    int iy = oy - 2 + gy, ix = ox - 2 + gx;
    iy = iy < 0 ? -iy : (iy > IMG_H - 1 ? 2 * (IMG_H - 1) - iy : iy);
    ix = ix < 0 ? -ix : (ix > IMG_W - 1 ? 2 * (IMG_W - 1) - ix : ix);
    long long o = (long long)iy * IMG_W + ix;
    float r = __builtin_nontemporal_load(&img[o]);
    float g = __builtin_nontemporal_load(&img[HW + o]);
    float b = __builtin_nontemporal_load(&img[2 * HW + o]);
    sg[i] = (0.299f * r + 0.587f * g + 0.114f * b) * 255.0f;
  }
  __syncthreads();

  // Stage B: |gx|+|gy| on 34x34 (zero outside image for NMS neighbor pad).
  for (int i = tid; i < 34 * 34; i += 256) {
    int my = i / 34, mx = i % 34;
    int iy = oy - 1 + my, ix = ox - 1 + mx;
    float m = 0.0f;
    if (iy >= 0 && iy < IMG_H && ix >= 0 && ix < IMG_W) {
      const float* c = &sg[(my + 1) * 36 + (mx + 1)];
      float gxv = (c[-36 + 1] + 2.0f * c[1] + c[36 + 1]) -
                  (c[-36 - 1] + 2.0f * c[-1] + c[36 - 1]);
      float gyv = (c[36 - 1] + 2.0f * c[36] + c[36 + 1]) -
                  (c[-36 - 1] + 2.0f * c[-36] + c[-36 + 1]);
      m = fabsf(gxv) + fabsf(gyv);
    }
    sm[i] = m;
  }
  __syncthreads();

  // Stage C: NMS + thresholds; 4 consecutive pixels -> one u32 mask store.
  {
    int cy = tid >> 3;              // 0..31
    int cx0 = (tid & 7) * 4;        // 0,4,...,28
    unsigned sOut = 0u, wOut = 0u;
    for (int k = 0; k < 4; ++k) {
      int cx = cx0 + k;
      const float* c = &sg[(cy + 2) * 36 + (cx + 2)];
      float gxv = (c[-36 + 1] + 2.0f * c[1] + c[36 + 1]) -
                  (c[-36 - 1] + 2.0f * c[-1] + c[36 - 1]);
      float gyv = (c[36 - 1] + 2.0f * c[36] + c[36 + 1]) -
                  (c[-36 - 1] + 2.0f * c[-36] + c[-36 + 1]);
      const float* mc = &sm[(cy + 1) * 34 + (cx + 1)];
      float mag = mc[0];
      float ax = fabsf(gxv), ay = fabsf(gyv);
      bool b0 = ay < 0.41421356237309503f * ax;   // deg in [0,22.5)|[157.5,180)
      bool b2 = ay >= 2.41421356237309515f * ax;  // deg in [67.5,112.5)
      bool pos = (gxv >= 0.0f) == (gyv >= 0.0f);  // slope sign
      bool b1 = !b0 && !b2 && pos;                // deg in [22.5,67.5)
      float n1, n2;
      if (b0)      { n1 = mc[1];       n2 = mc[-1]; }
      else if (b2) { n1 = mc[-34];     n2 = mc[34]; }
      else if (b1) { n1 = mc[-34 + 1]; n2 = mc[34 - 1]; }
      else         { n1 = mc[-34 - 1]; n2 = mc[34 + 1]; }
      float nms = ((mag >= n1) && (mag > n2)) ? mag : 0.0f;
      if (nms > 200.0f) sOut |= 0xFFu << (8 * k);
      if (nms > 100.0f) wOut |= 0xFFu << (8 * k);
    }
    long long p = (long long)(PAD + oy + cy) * PP + (PAD + ox + cx0);
    *(unsigned*)(edgesPad + p) = sOut;
    *(unsigned*)(weakPad + p)  = wOut;
  }
}

// --- Kernel 2: 10 hysteresis dilation steps per launch.
// 64x64 output tile + halo 10 -> 84x84 byte tiles DMA'd into LDS by the TDM.
// Masks are 0x00/0xFF bytes; dilate+min(weak) == 3x3 OR then AND, done 4
// pixels at a time on u32 words (cross-byte shifts supply x-neighbors).
#define TW     84
#define TWU    21                 // u32 words per tile row
#define TCELLS (TW * TWU)         // 1764 words per tile
__global__ __launch_bounds__(256) void canny_dilate_kernel(
    const unsigned char* __restrict__ src,
    unsigned char* __restrict__ dst,
    const unsigned char* __restrict__ weak) {
  __shared__ unsigned smem[3 * TCELLS + 4];  // [wk][eA][eB][slack]
  unsigned* wk = smem;
  unsigned* eA = smem + TCELLS;
  unsigned* eB = smem + 2 * TCELLS;
  const int tid = threadIdx.x;
  const int px0 = PAD - 10 + blockIdx.x * 64;   // >= 6, tile fits in padding
  const int py0 = PAD - 10 + blockIdx.y * 64;
  const long long off = (long long)py0 * PP + px0;

  if (tid < 32) {  // wave 0 issues both tensor DMAs, waits on TENSORcnt
    unsigned lbase = (unsigned)(uintptr_t)(void*)smem;
    tdm_load_tile_2d(lbase, weak, off, TW, TW);
    tdm_load_tile_2d(lbase + 4u * TCELLS, src, off, TW, TW);
    __builtin_amdgcn_s_wait_tensorcnt(0);
  }
  __syncthreads();

  unsigned* pr = eA;
  unsigned* nx = eB;
  for (int s = 0; s < 10; ++s) {
    for (int i = tid; i < 82 * TWU; i += 256) {
      int r = i / TWU + 1;                       // rows 1..82
      int b = r * TWU + (i % TWU);
      unsigned V  = pr[b - TWU]     | pr[b]     | pr[b + TWU];
      unsigned VL = pr[b - TWU - 1] | pr[b - 1] | pr[b + TWU - 1];
      unsigned VR = pr[b - TWU + 1] | pr[b + 1] | pr[b + TWU + 1];
      unsigned d = V | (V << 8) | (V >> 8) | (VL >> 24) | (VR << 24);
      nx[b] = d & wk[b];
    }
    __syncthreads();
    unsigned* t = pr; pr = nx; nx = t;           // after 10 steps: pr == eA
  }

  // Store the exact 64x64 interior (LDS coords [10,74)) as aligned u32s.
  for (int i = tid; i < 64 * 16; i += 256) {
    int r = i >> 4;                              // 0..63
    int c4 = i & 15;                             // u32 within 64-byte row
    int wb = (10 + r) * TWU + 2 + c4;            // byte 10+4*c4 -> shift 16
    unsigned out = (pr[wb] >> 16) | (pr[wb + 1] << 16);
    long long p = (long long)(PAD + blockIdx.y * 64 + r) * PP +
                  (PAD + blockIdx.x * 64 + 4 * c4);
    *(unsigned*)(dst + p) = out;
  }
}

// --- Kernel 3: edges byte mask -> 0.0/1.0 broadcast to 3 fp32 planes.
// 4 pixels/thread: one aligned u32 mask load, three non-temporal B128 stores
// (out is written exactly once; keep the 48MB stream out of L2).
__global__ void canny_final_kernel(const unsigned char* __restrict__ edgesPad,
                                   float* __restrict__ out) {
  const long long HW = (long long)IMG_H * IMG_W;
  const long long NV = HW >> 2;                  // 4-pixel groups
  long long i = (long long)blockIdx.x * blockDim.x + threadIdx.x;
  long long stride = (long long)gridDim.x * blockDim.x;
  for (; i < NV; i += stride) {
    long long pix = i << 2;
    long long y = pix >> 11;                     // /2048
    long long x = pix & (IMG_W - 1);             // multiple of 4
    unsigned m = *(const unsigned*)(edgesPad + (PAD + y) * PP + PAD + x);
    v4f v;
    v.x = (m & 0x000000FFu) ? 1.0f : 0.0f;
    v.y = (m & 0x0000FF00u) ? 1.0f : 0.0f;
    v.z = (m & 0x00FF0000u) ? 1.0f : 0.0f;
    v.w = (m & 0xFF000000u) ? 1.0f : 0.0f;
    __builtin_nontemporal_store(v, (v4f*)(out + pix));
    __builtin_nontemporal_store(v, (v4f*)(out + HW + pix));
    __builtin_nontemporal_store(v, (v4f*)(out + 2 * HW + pix));
  }
}

extern "C" void kernel_launch(void* const* d_in, const int* in_sizes, int n_in,
                              void* d_out, int out_size, void* d_ws, size_t ws_size,
                              hipStream_t stream) {
  (void)in_sizes; (void)n_in; (void)out_size; (void)ws_size;
  const float* img = (const float*)d_in[0];
  float* out = (float*)d_out;

  unsigned char* weakPad = (unsigned char*)d_ws;
  unsigned char* e0Pad   = weakPad + PBYTES;
  unsigned char* e1Pad   = weakPad + 2 * PBYTES;

  // 0) Zero all three padded mask buffers (borders stay 0 all pipeline long).
  {
    int nvec = (int)(3 * PBYTES / 16);
    canny_init_kernel<<<4096, 256, 0, stream>>>((v4u*)d_ws, nvec);
  }
  // 1) Gray + Sobel + NMS + double threshold -> strong(e0), weak masks.
  {
    dim3 grid(IMG_W / 32, IMG_H / 32);
    canny_thresh_kernel<<<grid, 256, 0, stream>>>(img, e0Pad, weakPad);
  }
  // 2) Hysteresis: exactly 100 monotone dilation steps (10 launches x 10
  //    in-LDS steps). Monotone + idempotent past fixed point == reference's
  //    capped while_loop. Ping-pong global buffers to avoid cross-block races.
  {
    dim3 grid(IMG_W / 64, IMG_H / 64);
    for (int i = 0; i < 10; ++i) {
      const unsigned char* src = (i & 1) ? e1Pad : e0Pad;
      unsigned char* dst       = (i & 1) ? e0Pad : e1Pad;
      canny_dilate_kernel<<<grid, 256, 0, stream>>>(src, dst, weakPad);
    }
    // i=9 wrote e0Pad -> final edges live in e0Pad.
  }
  // 3) Broadcast edges to 3 fp32 channels.
  canny_final_kernel<<<4096, 256, 0, stream>>>(e0Pad, out);
}